// S4_43885975830900
// MI455X (gfx1250) — compile-verified
//
#include <hip/hip_runtime.h>
#include <math.h>
#include <stdint.h>

// ---------------- problem sizes ----------------
constexpr int kB   = 16;    // batch
constexpr int kL   = 4096;  // sequence length
constexpr int kH   = 128;   // channels
constexpr int kN   = 64;    // state dim
constexpr int kOut = 128;   // output dim
constexpr int kT   = 64;    // chunk length
constexpr int kNC  = kL / kT;

// ---------------- workspace layout (floats), total ~39.9 MB ----------------
constexpr size_t M64_OFF = 0;                                  // A_bar^64   [H][N][N]
constexpr size_t F_OFF   = M64_OFF + (size_t)kH * kN * kN;     // F          [H][N][T]
constexpr size_t G_OFF   = F_OFF   + (size_t)kH * kN * kT;     // G          [H][T][N]
constexpr size_t K_OFF   = G_OFF   + (size_t)kH * kT * kN;     // taps       [H][T]
constexpr size_t Y_OFF   = K_OFF   + (size_t)kH * kT;          // yhat       [B*L][H]

typedef __attribute__((ext_vector_type(2))) float v2f;
typedef __attribute__((ext_vector_type(8))) float v8f;

static __device__ __forceinline__ v8f wmma4(v2f a, v2f b, v8f c) {
  // D = A(16x4 f32) * B(4x16 f32) + C(16x16 f32)
  return __builtin_amdgcn_wmma_f32_16x16x4_f32(false, a, false, b, (short)0, c,
                                               false, false);
}

// low 32 bits of a generic pointer to __shared__ == wave-relative LDS offset
static __device__ __forceinline__ uint32_t lds_off(const void* p) {
  return (uint32_t)(uintptr_t)p;
}

// =====================================================================
// Kernel 1: per-channel discretization + chunk operators (128 WGs).
//   Gauss-Jordan solve of (I - dt/2 A)[A_bar|B_bar] = [I + dt/2 A | dt*B],
//   Krylov vectors for F/G/taps, A^64 by 6 LDS squarings.
// =====================================================================
__global__ __launch_bounds__(256) void s4_precompute(
    const float* __restrict__ A, const float* __restrict__ Bm,
    const float* __restrict__ Cm, const float* __restrict__ log_dt,
    float* __restrict__ ws) {
  __shared__ float smem[64 * 130];   // GJ augmented; later 2x(64x65) ping-pong
  __shared__ float vv[kN], wv[kN], fac[kN];

  const int h   = blockIdx.x;
  const int tid = threadIdx.x;
  const float dt = expf(log_dt[h]);

#define AUG(r, c) smem[(r) * 130 + (c)]
  for (int idx = tid; idx < kN * 129; idx += 256) {
    int r = idx / 129, c = idx % 129;
    float v;
    if (c < kN)            v = (r == c ? 1.f : 0.f) - 0.5f * dt * A[r * kN + c];
    else if (c < 2 * kN)   v = (r == c - kN ? 1.f : 0.f) + 0.5f * dt * A[r * kN + (c - kN)];
    else                   v = dt * Bm[h * kN + r];
    AUG(r, c) = v;
  }
  __syncthreads();

  // Gauss-Jordan; lhs lower-triangular with diag = 1 + dt/2*(i+1) -> no pivoting.
  for (int p = 0; p < kN; ++p) {
    const float inv = 1.0f / AUG(p, p);
    __syncthreads();  // all read pivot before row p is scaled
    for (int c = tid; c < 129; c += 256) AUG(p, c) *= inv;
    __syncthreads();
    if (tid < kN) fac[tid] = (tid == p) ? 0.f : AUG(tid, p);
    __syncthreads();
    for (int idx = tid; idx < kN * 129; idx += 256) {
      int r = idx / 129, c = idx % 129;
      AUG(r, c) -= fac[r] * AUG(p, c);
    }
    __syncthreads();
  }

  // Repack A_bar into a 64x65-padded buffer (via registers: regions overlap).
  float tA[16];
  for (int t = 0; t < 16; ++t) {
    int idx = tid + 256 * t;  // 0..4095
    tA[t] = AUG(idx >> 6, 64 + (idx & 63));
  }
  const float bbar = (tid < kN) ? AUG(tid, 128) : 0.f;
  const float cval = (tid < kN) ? Cm[h * kN + tid] : 0.f;
  __syncthreads();
#define AB(r, c) smem[(r) * 65 + (c)]
#define M2(r, c) smem[64 * 65 + (r) * 65 + (c)]
  for (int t = 0; t < 16; ++t) {
    int idx = tid + 256 * t;
    AB(idx >> 6, idx & 63) = tA[t];
  }
  if (tid < kN) { vv[tid] = bbar; wv[tid] = cval; }
  __syncthreads();

  // Krylov: v_m = A_bar^m B_bar, w_m = C A_bar^m.
  //   taps k[m] = C.v_m ; F[:, T-1-m] = v_m ; G[m,:] = w_{m+1}.
  float* Fp = ws + F_OFF + (size_t)h * kN * kT;
  float* Gp = ws + G_OFF + (size_t)h * kT * kN;
  float* Kp = ws + K_OFF + (size_t)h * kT;
  for (int m = 0; m < kT; ++m) {
    if (tid < kN) Fp[tid * kT + (kT - 1 - m)] = vv[tid];
    if (tid == 0) {
      float s = 0.f;
      for (int n = 0; n < kN; ++n) s += Cm[h * kN + n] * vv[n];
      Kp[m] = s;
    }
    float nv = 0.f, nw = 0.f;
    if (tid < kN) {
      for (int j = 0; j < kN; ++j) {
        nv += AB(tid, j) * vv[j];
        nw += wv[j] * AB(j, tid);
      }
    }
    __syncthreads();
    if (tid < kN) {
      vv[tid] = nv;
      wv[tid] = nw;
      Gp[m * kN + tid] = nw;
    }
    __syncthreads();
  }

  // A^64 by repeated squaring (6x); even count -> result lands in AB region.
  for (int sq = 0; sq < 6; ++sq) {
    const bool srcIsAB = ((sq & 1) == 0);
    for (int idx = tid; idx < kN * kN; idx += 256) {
      int r = idx >> 6, c = idx & 63;
      float s = 0.f;
      if (srcIsAB) {
        for (int j = 0; j < kN; ++j) s += AB(r, j) * AB(j, c);
        M2(r, c) = s;
      } else {
        for (int j = 0; j < kN; ++j) s += M2(r, j) * M2(j, c);
        AB(r, c) = s;
      }
    }
    __syncthreads();
  }
  float* Mp = ws + M64_OFF + (size_t)h * kN * kN;
  for (int idx = tid; idx < kN * kN; idx += 256)
    Mp[idx] = AB(idx >> 6, idx & 63);
#undef AUG
#undef AB
#undef M2
}

// =====================================================================
// Kernel 2: chunk-serial scan, one WG per channel, 4 waves.
//   Per chunk: Y = tril_toeplitz(k) @ X + G @ S + D*x ; S = A64 @ S + F @ X.
//   All four products are 64xKx16 fp32 WMMA matmuls.
//   X is staged via double-buffered async global->LDS loads (ASYNCcnt):
//   chunk c+1's gather overlaps chunk c's WMMA work.
// =====================================================================
__global__ __launch_bounds__(128) void s4_scan(const float* __restrict__ x,
                                               const float* __restrict__ Dv,
                                               float* __restrict__ ws) {
  __shared__ float sM[kN][kN + 1];
  __shared__ float sF[kN][kT + 1];
  __shared__ float sG[kT][kN + 1];
  __shared__ float sKtap[kT];
  __shared__ float sX[2][kT][kB + 1];   // double buffer for async staging
  __shared__ float sS[kN][kB + 1];

  const int h     = blockIdx.x;
  const int tid   = threadIdx.x;
  const int lane  = tid & 31;
  const int lhalf = lane & 15;   // N (col) index / M index within tile
  const int khi   = lane >> 4;   // which K pair / which row-half of D
  const int m0    = (tid >> 5) * 16;  // wave's 16-row strip

  const float* Mp = ws + M64_OFF + (size_t)h * kN * kN;
  const float* Fp = ws + F_OFF + (size_t)h * kN * kT;
  const float* Gp = ws + G_OFF + (size_t)h * kT * kN;
  const float* Kp = ws + K_OFF + (size_t)h * kT;
  float* yhat = ws + Y_OFF;

  // async-gather one chunk of x into sX[chunk&1]; 8 ops per wave, EXEC full.
  auto issueX = [&](int chunk) {
    const int buf = chunk & 1;
    const int t0 = chunk * kT;
    for (int idx = tid; idx < kT * kB; idx += 128) {
      int i = idx >> 4, bb = idx & 15;
      uint32_t doff = lds_off(&sX[buf][i][bb]);
      const float* gp = &x[((size_t)bb * kL + t0 + i) * kH + h];
      asm volatile("global_load_async_to_lds_b32 %0, %1, off"
                   :: "v"(doff), "v"(gp) : "memory");
    }
  };

  for (int idx = tid; idx < kN * kN; idx += 128) {
    int r = idx >> 6, c = idx & 63;
    sM[r][c] = Mp[idx];
    sF[r][c] = Fp[idx];
    sG[r][c] = Gp[idx];
  }
  if (tid < kT) sKtap[tid] = Kp[tid];
  for (int idx = tid; idx < kN * kB; idx += 128) sS[idx >> 4][idx & 15] = 0.f;
  const float dskip = Dv[h];
  issueX(0);

  const int ar = m0 + lhalf;  // A-fragment row for this lane
  for (int c = 0; c < kNC; ++c) {
    const int t0 = c * kT;
    const int xb = c & 1;
    if (c + 1 < kNC) {
      issueX(c + 1);  // overlap next gather with this chunk's WMMAs
      // async loads retire in order: <=8 outstanding => chunk c's 8 are done
      asm volatile("s_wait_asynccnt 8" ::: "memory");
    } else {
      asm volatile("s_wait_asynccnt 0" ::: "memory");
    }
    __syncthreads();

    // Y(64x16) = Toeplitz(k) @ X + G @ S
    v8f accY = {};
    for (int k0 = 0; k0 < kT; k0 += 4) {
      const int kc = k0 + 2 * khi;
      v2f a, b;
      a[0] = (ar >= kc) ? sKtap[ar - kc] : 0.f;
      a[1] = (ar >= kc + 1) ? sKtap[ar - kc - 1] : 0.f;
      b[0] = sX[xb][kc][lhalf];
      b[1] = sX[xb][kc + 1][lhalf];
      accY = wmma4(a, b, accY);
      a[0] = sG[ar][kc];
      a[1] = sG[ar][kc + 1];
      b[0] = sS[kc][lhalf];
      b[1] = sS[kc + 1][lhalf];
      accY = wmma4(a, b, accY);
    }
    // S'(64x16) = A64 @ S + F @ X
    v8f accS = {};
    for (int k0 = 0; k0 < kT; k0 += 4) {
      const int kc = k0 + 2 * khi;
      v2f a, b;
      a[0] = sM[ar][kc];
      a[1] = sM[ar][kc + 1];
      b[0] = sS[kc][lhalf];
      b[1] = sS[kc + 1][lhalf];
      accS = wmma4(a, b, accS);
      a[0] = sF[ar][kc];
      a[1] = sF[ar][kc + 1];
      b[0] = sX[xb][kc][lhalf];
      b[1] = sX[xb][kc + 1][lhalf];
      accS = wmma4(a, b, accS);
    }
    __syncthreads();  // all waves done reading old S
#pragma unroll
    for (int i = 0; i < 8; ++i) sS[m0 + i + khi * 8][lhalf] = accS[i];
#pragma unroll
    for (int i = 0; i < 8; ++i) {
      const int tt = m0 + i + khi * 8;
      const float yv = accY[i] + dskip * sX[xb][tt][lhalf];
      yhat[((size_t)lhalf * kL + t0 + tt) * kH + h] = yv;
    }
    __syncthreads();
  }
}

// =====================================================================
// Kernel 3: out = yhat @ W^T + bias.  (B*L x H) @ (H x OUT), fp32 WMMA.
//   64-row tile per WG, staged via async global->LDS B128 copies;
//   W staged transposed in LDS in K-slices of 32.
// =====================================================================
constexpr int kRowP = kH + 4;  // 132: 16B-aligned rows, bank shift of 4/row

__global__ __launch_bounds__(256) void s4_proj(const float* __restrict__ yhat,
                                               const float* __restrict__ W,
                                               const float* __restrict__ bias,
                                               float* __restrict__ out) {
  __shared__ float sA[64 * kRowP];       // yhat tile (async-filled)
  __shared__ float sW[32][kOut + 1];     // k-slice of W^T: sW[k][o] = W[o][kb+k]

  const int tid   = threadIdx.x;
  const size_t m0 = (size_t)blockIdx.x * 64;
  const int lane  = tid & 31;
  const int lhalf = lane & 15;
  const int khi   = lane >> 4;
  const int wave  = tid >> 5;
  const int rm    = (wave & 3) * 16;      // row strip
  const int oc0   = (wave >> 2) * 64;     // column half

  // async copy: 64 rows x 32 segments of 16B; 8 ops per wave, EXEC full
  for (int seg = tid; seg < 64 * (kH / 4); seg += 256) {
    int r = seg >> 5, cs = seg & 31;
    uint32_t doff = lds_off(&sA[r * kRowP + cs * 4]);
    const float* gp = &yhat[(m0 + r) * kH + cs * 4];
    asm volatile("global_load_async_to_lds_b128 %0, %1, off"
                 :: "v"(doff), "v"(gp) : "memory");
  }
  asm volatile("s_wait_asynccnt 0" ::: "memory");

  v8f acc[4] = {};
  for (int kb = 0; kb < kH; kb += 32) {
    __syncthreads();  // previous slice fully consumed (also fences sA asyncs)
    for (int idx = tid; idx < 32 * kOut; idx += 256) {
      int k = idx >> 7, o = idx & 127;
      sW[k][o] = W[(size_t)o * kH + kb + k];
    }
    __syncthreads();
    for (int kk = 0; kk < 32; kk += 4) {
      const int kc = kk + 2 * khi;
      v2f a;
      a[0] = sA[(rm + lhalf) * kRowP + kb + kc];
      a[1] = sA[(rm + lhalf) * kRowP + kb + kc + 1];
#pragma unroll
      for (int t = 0; t < 4; ++t) {
        v2f b;
        b[0] = sW[kc][oc0 + t * 16 + lhalf];
        b[1] = sW[kc + 1][oc0 + t * 16 + lhalf];
        acc[t] = wmma4(a, b, acc[t]);
      }
    }
  }
#pragma unroll
  for (int t = 0; t < 4; ++t) {
    const int o = oc0 + t * 16 + lhalf;
    const float bv = bias[o];
#pragma unroll
    for (int i = 0; i < 8; ++i) {
      const size_t row = m0 + rm + i + khi * 8;
      out[row * kOut + o] = acc[t][i] + bv;
    }
  }
}

// =====================================================================
extern "C" void kernel_launch(void* const* d_in, const int* in_sizes, int n_in,
                              void* d_out, int out_size, void* d_ws,
                              size_t ws_size, hipStream_t stream) {
  const float* x      = (const float*)d_in[0];
  const float* A      = (const float*)d_in[1];
  const float* Bm     = (const float*)d_in[2];
  const float* Cm     = (const float*)d_in[3];
  const float* Dv     = (const float*)d_in[4];
  const float* log_dt = (const float*)d_in[5];
  const float* W      = (const float*)d_in[6];
  const float* bias   = (const float*)d_in[7];
  float* out = (float*)d_out;
  float* ws  = (float*)d_ws;  // requires ~40 MB scratch (see layout above)

  s4_precompute<<<kH, 256, 0, stream>>>(A, Bm, Cm, log_dt, ws);
  s4_scan<<<kH, 128, 0, stream>>>(x, Dv, ws);
  s4_proj<<<(kB * kL) / 64, 256, 0, stream>>>(ws + Y_OFF, W, bias, out);
}